// GAT_66537633350226
// MI455X (gfx1250) — compile-verified
//
#include <hip/hip_runtime.h>
#include <hip/hip_bf16.h>

// ---------------------------------------------------------------------------
// GAT attention-weights kernel for MI455X (gfx1250).
// N=4096, F_IN=256, D_OUT=128, E=N*N/2 per graph, 2 graphs (block-diagonal).
// Memory-bound: 268MB int64 edge list per pass + 67MB f32 output.
// NT hints keep the 67MB intermediate resident in the 192MB L2 between
// edge_sum and edge_scale.
// ---------------------------------------------------------------------------

#define GAT_N      4096
#define GAT_F      256
#define GAT_D      128
#define GAT_E      8388608LL          // N*N/2 edges per graph
#define NODES      8192               // 2*N
#define SEG_LOC    4096               // segments touched per graph
#define EDGE_BLKS  1024               // 512 per graph
#define PAIRS_PB   8192               // 2-edge pairs per block
#define INT_NEG_INF (-2147483647 - 1)

typedef __attribute__((ext_vector_type(2))) float     v2f;
typedef __attribute__((ext_vector_type(8))) float     v8f;
typedef __attribute__((ext_vector_type(2))) long long v2ll;

// Monotonic int encoding of f32 (so integer atomicMax == float max).
__device__ __forceinline__ int enc_f32(float x) {
    int i = __float_as_int(x);
    return (i >= 0) ? i : (i ^ 0x7fffffff);
}
__device__ __forceinline__ float dec_f32(int i) {
    return __int_as_float((i >= 0) ? i : (i ^ 0x7fffffff));
}
__device__ __forceinline__ float lrelu(float t) {
    return t > 0.0f ? t : 0.2f * t;
}

// ---------------------------------------------------------------------------
// 0) Fold W into w_s = W@att_src, w_d = W@att_dst, and build the zero-padded
//    column-major WMMA B buffer: Bbuf[n*256 + k] (col 0 = w_s, col 1 = w_d).
// ---------------------------------------------------------------------------
__global__ void prep_wsd(const float* __restrict__ W,
                         const float* __restrict__ att_src,
                         const float* __restrict__ att_dst,
                         float* __restrict__ bbuf) {
    int j = threadIdx.x;                       // 0..255 (F_IN)
    float s = 0.f, d = 0.f;
    const float* row = W + j * GAT_D;
    #pragma unroll 8
    for (int t = 0; t < GAT_D; ++t) {
        float w = row[t];
        s += w * att_src[t];
        d += w * att_dst[t];
    }
    bbuf[0 * GAT_F + j] = s;                   // col 0: w_s
    bbuf[1 * GAT_F + j] = d;                   // col 1: w_d
    #pragma unroll
    for (int n = 2; n < 16; ++n) bbuf[n * GAT_F + j] = 0.0f;
}

// ---------------------------------------------------------------------------
// 1) Re-init segment accumulators every launch (deterministic replays).
// ---------------------------------------------------------------------------
__global__ void init_seg(int* __restrict__ segmax, float* __restrict__ denom) {
    int i = blockIdx.x * blockDim.x + threadIdx.x;
    if (i < NODES) { segmax[i] = INT_NEG_INF; denom[i] = 0.0f; }
}

// ---------------------------------------------------------------------------
// 2) a_s / a_d via fp32 WMMA: per wave, a 16-row tile of x (concat x1|x2)
//    times B[256,16] (col 0 = w_s, col 1 = w_d, rest zero; from bbuf).
//    A (16x4 f32): lane L holds M=L&15, K = (L>>4)*2 + {0,1}.
//    B (4x16 f32): lane L holds N=L&15, K = (L>>4)*2 + {0,1}.
//    D (16x16 f32): lane L (L&15 == N), vgpr r -> M = r + 8*(L>>4).
// ---------------------------------------------------------------------------
__global__ __launch_bounds__(32)
void wmma_as_ad(const float* __restrict__ x1, const float* __restrict__ x2,
                const float* __restrict__ bbuf,
                float* __restrict__ a_s, float* __restrict__ a_d) {
    const int tile = blockIdx.x;               // 512 tiles of 16 rows
    const int lane = threadIdx.x;              // wave32
    const int row0 = tile * 16;
    const int m    = lane & 15;
    const int kb   = (lane >> 4) * 2;

    const int grow = row0 + m;
    const float* xr = (grow < GAT_N) ? (x1 + (size_t)grow * GAT_F)
                                     : (x2 + (size_t)(grow - GAT_N) * GAT_F);
    const float* bc = bbuf + (size_t)(lane & 15) * GAT_F;   // column for this lane

    v8f c = {};
    #pragma unroll 4
    for (int k0 = 0; k0 < GAT_F; k0 += 4) {
        v2f a = *(const v2f*)(xr + k0 + kb);
        v2f b = *(const v2f*)(bc + k0 + kb);
        c = __builtin_amdgcn_wmma_f32_16x16x4_f32(
                false, a, false, b, (short)0, c, false, false);
    }

    const int r8 = (lane >> 4) * 8;
    if ((lane & 15) == 0) {
        #pragma unroll
        for (int r = 0; r < 8; ++r) a_s[row0 + r8 + r] = c[r];
    } else if ((lane & 15) == 1) {
        #pragma unroll
        for (int r = 0; r < 8; ++r) a_d[row0 + r8 + r] = c[r];
    }
}

// ---------------------------------------------------------------------------
// 3) Segment max over destination nodes, staged in LDS.
//    Blocks [0,512) -> graph1, [512,1024) -> graph2 (ids shifted by N).
//    Index streams are non-temporal (one-shot, keep L2 for hot tables).
// ---------------------------------------------------------------------------
__global__ __launch_bounds__(256)
void edge_max(const long long* __restrict__ ei1,
              const long long* __restrict__ ei2,
              const float* __restrict__ a_s, const float* __restrict__ a_d,
              int* __restrict__ segmax) {
    __shared__ int lmax[SEG_LOC];
    for (int i = threadIdx.x; i < SEG_LOC; i += 256) lmax[i] = INT_NEG_INF;
    __syncthreads();

    const int b   = blockIdx.x;
    const int g2  = (b >= 512) ? 1 : 0;
    const int off = g2 * GAT_N;
    const long long* src = g2 ? ei2 : ei1;     // row 0: src, row 1: dst
    const long long* dst = src + GAT_E;
    const long long pairBase = (long long)(b & 511) * PAIRS_PB;

    const v2ll* s2 = (const v2ll*)src;
    const v2ll* d2 = (const v2ll*)dst;

    for (int it = 0; it < 32; ++it) {
        long long p = pairBase + it * 256 + threadIdx.x;
        v2ll sp = __builtin_nontemporal_load(s2 + p);
        v2ll dp = __builtin_nontemporal_load(d2 + p);
        int s0 = (int)sp.x, s1 = (int)sp.y;
        int dd0 = (int)dp.x, dd1 = (int)dp.y;
        float e0 = lrelu(a_s[s0 + off] + a_d[dd0 + off]);
        float e1 = lrelu(a_s[s1 + off] + a_d[dd1 + off]);
        atomicMax(&lmax[dd0], enc_f32(e0));
        atomicMax(&lmax[dd1], enc_f32(e1));
    }
    __syncthreads();
    for (int i = threadIdx.x; i < SEG_LOC; i += 256) {
        int v = lmax[i];
        if (v != INT_NEG_INF) atomicMax(&segmax[i + off], v);
    }
}

// ---------------------------------------------------------------------------
// 4) ex = exp(e - max[dst]) -> d_out[k] (RT store: want L2 residency for the
//    re-read in edge_scale); denominators staged in LDS.
// ---------------------------------------------------------------------------
__global__ __launch_bounds__(256)
void edge_sum(const long long* __restrict__ ei1,
              const long long* __restrict__ ei2,
              const float* __restrict__ a_s, const float* __restrict__ a_d,
              const int* __restrict__ segmax, float* __restrict__ denom,
              float* __restrict__ out) {
    __shared__ float smaxf[SEG_LOC];
    __shared__ float lden[SEG_LOC];
    const int b   = blockIdx.x;
    const int g2  = (b >= 512) ? 1 : 0;
    const int off = g2 * GAT_N;

    for (int i = threadIdx.x; i < SEG_LOC; i += 256) {
        smaxf[i] = dec_f32(segmax[i + off]);
        lden[i]  = 0.0f;
    }
    __syncthreads();

    const long long* src = g2 ? ei2 : ei1;
    const long long* dst = src + GAT_E;
    const long long pairBase = (long long)(b & 511) * PAIRS_PB;
    const v2ll* s2 = (const v2ll*)src;
    const v2ll* d2 = (const v2ll*)dst;
    v2f* o2 = (v2f*)(out + (long long)g2 * GAT_E);

    for (int it = 0; it < 32; ++it) {
        long long p = pairBase + it * 256 + threadIdx.x;
        v2ll sp = __builtin_nontemporal_load(s2 + p);
        v2ll dp = __builtin_nontemporal_load(d2 + p);
        int s0 = (int)sp.x, s1 = (int)sp.y;
        int dd0 = (int)dp.x, dd1 = (int)dp.y;
        float e0 = lrelu(a_s[s0 + off] + a_d[dd0 + off]);
        float e1 = lrelu(a_s[s1 + off] + a_d[dd1 + off]);
        float x0 = __expf(e0 - smaxf[dd0]);
        float x1 = __expf(e1 - smaxf[dd1]);
        atomicAdd(&lden[dd0], x0);
        atomicAdd(&lden[dd1], x1);
        v2f o; o.x = x0; o.y = x1;
        o2[p] = o;                              // RT: stay in L2 for next pass
    }
    __syncthreads();
    for (int i = threadIdx.x; i < SEG_LOC; i += 256) {
        float v = lden[i];
        if (v != 0.0f) atomicAdd(&denom[i + off], v);
    }
}

// ---------------------------------------------------------------------------
// 5) alpha = ex / denom[dst]; dst row NT-read, ex read hits L2, final store NT.
// ---------------------------------------------------------------------------
__global__ __launch_bounds__(256)
void edge_scale(const long long* __restrict__ ei1,
                const long long* __restrict__ ei2,
                const float* __restrict__ denom, float* __restrict__ out) {
    __shared__ float rden[SEG_LOC];
    const int b   = blockIdx.x;
    const int g2  = (b >= 512) ? 1 : 0;
    const int off = g2 * GAT_N;

    for (int i = threadIdx.x; i < SEG_LOC; i += 256) {
        float d = denom[i + off];
        rden[i] = (d != 0.0f) ? (1.0f / d) : 0.0f;
    }
    __syncthreads();

    const long long* src = g2 ? ei2 : ei1;
    const long long* dst = src + GAT_E;
    const long long pairBase = (long long)(b & 511) * PAIRS_PB;
    const v2ll* d2 = (const v2ll*)dst;
    v2f* o2 = (v2f*)(out + (long long)g2 * GAT_E);

    for (int it = 0; it < 32; ++it) {
        long long p = pairBase + it * 256 + threadIdx.x;
        v2ll dp = __builtin_nontemporal_load(d2 + p);
        v2f v = o2[p];                          // expected L2 hit
        v.x *= rden[(int)dp.x];
        v.y *= rden[(int)dp.y];
        __builtin_nontemporal_store(v, o2 + p); // final result: no re-read
    }
}

// ---------------------------------------------------------------------------
// Launcher
// ---------------------------------------------------------------------------
extern "C" void kernel_launch(void* const* d_in, const int* in_sizes, int n_in,
                              void* d_out, int out_size, void* d_ws, size_t ws_size,
                              hipStream_t stream) {
    const float*     x1      = (const float*)d_in[0];
    const float*     x2      = (const float*)d_in[1];
    const long long* ei1     = (const long long*)d_in[2];
    const long long* ei2     = (const long long*)d_in[3];
    const float*     W       = (const float*)d_in[4];
    const float*     att_src = (const float*)d_in[5];
    const float*     att_dst = (const float*)d_in[6];
    float*           out     = (float*)d_out;

    char* ws = (char*)d_ws;
    float* bbuf   = (float*)(ws);                          // 16*256 f32 (16KB)
    float* a_s    = (float*)(ws + 16384);                  // 8192 f32
    float* a_d    = (float*)(ws + 16384 + 32768);          // 8192 f32
    int*   segmax = (int*)  (ws + 16384 + 65536);          // 8192 i32 (encoded)
    float* denom  = (float*)(ws + 16384 + 65536 + 32768);  // 8192 f32

    prep_wsd  <<<1,   256, 0, stream>>>(W, att_src, att_dst, bbuf);
    init_seg  <<<32,  256, 0, stream>>>(segmax, denom);
    wmma_as_ad<<<NODES / 16, 32, 0, stream>>>(x1, x2, bbuf, a_s, a_d);
    edge_max  <<<EDGE_BLKS, 256, 0, stream>>>(ei1, ei2, a_s, a_d, segmax);
    edge_sum  <<<EDGE_BLKS, 256, 0, stream>>>(ei1, ei2, a_s, a_d, segmax, denom, out);
    edge_scale<<<EDGE_BLKS, 256, 0, stream>>>(ei1, ei2, denom, out);
}